// Local3dGeodesicConv_50019189129842
// MI455X (gfx1250) — compile-verified
//
#include <hip/hip_runtime.h>
#include <hip/hip_bf16.h>
#include <math.h>

typedef __attribute__((ext_vector_type(2))) float v2f;
typedef __attribute__((ext_vector_type(8))) float v8f;

#define NV     32768
#define NDIRS  16
#define NF     16
#define NRINGS 3
#define EPSN   1e-12f

// ---------------- per-batch mean of y (B blocks, 256 threads) ----------------
__global__ __launch_bounds__(256) void gc_mean_kernel(const float* __restrict__ y,
                                                      float* __restrict__ mean) {
  __shared__ float red[3][256];
  const int b = blockIdx.x;
  const int t = threadIdx.x;
  const float* yb = y + (size_t)b * NV * 3;
  float sx = 0.f, sy = 0.f, sz = 0.f;
  for (int v = t; v < NV; v += 256) {
    sx += yb[(size_t)v * 3 + 0];
    sy += yb[(size_t)v * 3 + 1];
    sz += yb[(size_t)v * 3 + 2];
  }
  red[0][t] = sx; red[1][t] = sy; red[2][t] = sz;
  __syncthreads();
  for (int s = 128; s > 0; s >>= 1) {
    if (t < s) {
      red[0][t] += red[0][t + s];
      red[1][t] += red[1][t + s];
      red[2][t] += red[2][t + s];
    }
    __syncthreads();
  }
  if (t == 0) {
    const float inv = 1.0f / (float)NV;
    mean[b * 3 + 0] = red[0][0] * inv;
    mean[b * 3 + 1] = red[1][0] * inv;
    mean[b * 3 + 2] = red[2][0] * inv;
  }
}

// ---------------- main kernel: 2 vertices per wave, WMMA 16x16x4 tail --------
__global__ __launch_bounds__(256) void gc_main_kernel(
    const float* __restrict__ y, const int* __restrict__ contrib,
    const float* __restrict__ wts, const float* __restrict__ pk,
    const float* __restrict__ bias, const float* __restrict__ mean,
    float* __restrict__ out) {
  const int lane = threadIdx.x & 31;
  const int wave = threadIdx.x >> 5;
  const int d    = lane & 15;          // direction index (and output column n)
  const bool lo  = lane < 16;          // lower half-wave = vertex 0

  // linear vertex ids: this wave owns v0 (lanes 0-15) and v1 (lanes 16-31)
  const long long vbase = (long long)blockIdx.x * 16 + (long long)wave * 2;
  const long long v = vbase + (lo ? 0 : 1);
  const int b = (int)(v >> 15);        // v / NV  (NV = 2^15)
  const float* yb = y + (size_t)b * NV * 3;

  // center point
  const float cx = y[(size_t)v * 3 + 0];
  const float cy = y[(size_t)v * 3 + 1];
  const float cz = y[(size_t)v * 3 + 2];

  // ring-0 weighted gather of 3 contributors for this direction
  const size_t cb = (size_t)v * (NRINGS * NDIRS * 3) + (size_t)d * 3;
  float wx = 0.f, wy = 0.f, wz = 0.f;
#pragma unroll
  for (int k = 0; k < 3; ++k) {
    const int   gi = contrib[cb + k];
    const float wk = wts[cb + k];
    const float* gp = yb + (size_t)gi * 3;
    wx = fmaf(wk, gp[0], wx);
    wy = fmaf(wk, gp[1], wy);
    wz = fmaf(wk, gp[2], wz);
  }

  // x = l2_normalize(w - center)
  float xx = wx - cx, xy = wy - cy, xz = wz - cz;
  {
    const float sq = xx * xx + xy * xy + xz * xz;
    const float r = 1.0f / sqrtf(fmaxf(sq, EPSN));
    xx *= r; xy *= r; xz *= r;
  }

  // x_prev = x from direction (d-1) mod 16, within the same half-wave
  const int prev = (lane & 16) | ((lane + 15) & 15);
  const float px_ = __shfl(xx, prev, 32);
  const float py_ = __shfl(xy, prev, 32);
  const float pz_ = __shfl(xz, prev, 32);

  // cross(x, x_prev), then sum over 16 directions (butterfly within half-wave)
  float zx = xy * pz_ - xz * py_;
  float zy = xz * px_ - xx * pz_;
  float zz = xx * py_ - xy * px_;
#pragma unroll
  for (int m = 1; m <= 8; m <<= 1) {
    zx += __shfl_xor(zx, m, 32);
    zy += __shfl_xor(zy, m, 32);
    zz += __shfl_xor(zz, m, 32);
  }
  {
    const float sq = zx * zx + zy * zy + zz * zz;
    const float r = 1.0f / sqrtf(fmaxf(sq, EPSN));
    zx *= r; zy *= r; zz *= r;
  }

  // x <- l2_normalize( x * (1 - x.z) )   (exactly as the reference computes)
  {
    const float dzx = xx * zx + xy * zy + xz * zz;
    const float s = 1.0f - dzx;
    float tx = xx * s, ty = xy * s, tz = xz * s;
    const float sq = tx * tx + ty * ty + tz * tz;
    const float r = 1.0f / sqrtf(fmaxf(sq, EPSN));
    xx = tx * r; xy = ty * r; xz = tz * r;
  }

  // yv = cross(z, x)
  const float yx = zy * xz - zz * xy;
  const float yy = zz * xx - zx * xz;
  const float yz = zx * xy - zy * xx;

  // pos = center - batch mean
  const float posx = cx - mean[b * 3 + 0];
  const float posy = cy - mean[b * 3 + 1];
  const float posz = cz - mean[b * 3 + 2];

  // the three dot products = A-matrix row entries (K = 0,1,2; K=3 is zero pad)
  const float d0 = xx * posx + xy * posy + xz * posz;  // x  . pos
  const float d1 = yx * posx + yy * posy + yz * posz;  // yv . pos
  const float d2 = zx * posx + zy * posy + zz * posz;  // z  . pos

  // ---- B matrix: position_kernel (3x16), zero-padded to 4x16 ----
  // 16x4-class f32 operand layout: VGPR0 = K0 (lanes 0-15) / K2 (lanes 16-31),
  //                                VGPR1 = K1 (lanes 0-15) / K3 (lanes 16-31)
  v2f Bm;
  Bm.x = lo ? pk[d] : pk[2 * NF + d];
  Bm.y = lo ? pk[NF + d] : 0.0f;

  // ---- C matrix: bias broadcast down each column (C[m,n] = bias[n]) ----
  const float bl = bias[d];
  v8f C = {bl, bl, bl, bl, bl, bl, bl, bl};

  // ---- A matrix for vertex 0 (rows M=0..15 live in lanes 0..15) ----
  const float s2 = __shfl(d2, d, 32);            // lane 16+r <- d2 of lane r
  v2f A0;
  A0.x = lo ? d0 : s2;                           // K0 / K2
  A0.y = lo ? d1 : 0.0f;                         // K1 / K3(pad)

  // ---- A matrix for vertex 1 (rows M=0..15 live in lanes 16..31) ----
  const float s0 = __shfl(d0, d | 16, 32);       // lane r <- d0 of lane 16+r
  const float s1 = __shfl(d1, d | 16, 32);
  v2f A1;
  A1.x = lo ? s0 : d2;
  A1.y = lo ? s1 : 0.0f;

  // D = A x B + C  (16x16 f32 accumulator, 8 VGPRs)
  v8f D0 = __builtin_amdgcn_wmma_f32_16x16x4_f32(false, A0, false, Bm,
                                                 (short)0, C, false, false);
  v8f D1 = __builtin_amdgcn_wmma_f32_16x16x4_f32(false, A1, false, Bm,
                                                 (short)0, C, false, false);

  // ---- ReLU + store. C/D layout: lane holds column n=d; VGPR j holds row
  //      m = j (lanes 0-15) or m = j+8 (lanes 16-31). out[v][m][n], n fastest.
  float* o0 = out + (size_t)vbase * (NDIRS * NF);
  float* o1 = o0 + NDIRS * NF;
  const int roff = lo ? 0 : 8;
#pragma unroll
  for (int j = 0; j < 8; ++j) {
    o0[(j + roff) * NF + d] = fmaxf(D0[j], 0.0f);
    o1[(j + roff) * NF + d] = fmaxf(D1[j], 0.0f);
  }
}

extern "C" void kernel_launch(void* const* d_in, const int* in_sizes, int n_in,
                              void* d_out, int out_size, void* d_ws, size_t ws_size,
                              hipStream_t stream) {
  const float* y       = (const float*)d_in[0];  // (B, NV, 3) f32
  const int*   contrib = (const int*)  d_in[1];  // (B, NV, 3, 16, 3) i32
  const float* wts     = (const float*)d_in[2];  // (B, NV, 3, 16, 3) f32
  const float* pk      = (const float*)d_in[3];  // (3, 16) f32
  const float* bias    = (const float*)d_in[4];  // (16,) f32
  float*       out     = (float*)d_out;          // (B, NV, 16, 16) f32
  float*       mean    = (float*)d_ws;           // B*3 floats of scratch

  const int B = in_sizes[0] / (NV * 3);

  gc_mean_kernel<<<B, 256, 0, stream>>>(y, mean);

  // 256 threads = 8 waves = 16 vertices per block; exact cover (B*NV % 16 == 0)
  const int blocks = (B * NV) / 16;
  gc_main_kernel<<<blocks, 256, 0, stream>>>(y, contrib, wts, pk, bias, mean, out);
}